// GraphSAGE_68882685493810
// MI455X (gfx1250) — compile-verified
//
#include <hip/hip_runtime.h>

typedef __attribute__((ext_vector_type(2))) float v2f;
typedef __attribute__((ext_vector_type(4))) float v4f;
typedef __attribute__((ext_vector_type(8))) float v8f;

#define D_IN   128     // per-layer feature width (D == OUT == 128)
#define K_NEI  16      // neighbors per node
#define KDIM   256     // GEMM K = 2*D (concat of self + mean)
#define NODES  32      // rows per block
#define CSTR   260     // comb row stride (floats): 260 % 64 == 4 -> a b64
                       // A-frag load touches all 64 banks exactly once; rows
                       // are 1040B (16B multiple) so b64/b128 stay aligned.

// ---------------------------------------------------------------------------
// Build pair-interleaved transposed weight:
//   WP[k][o] = ( W[o][k], W[o][(k+2) & 255] )   as float2, [256][128]
// One global_load_b64 then yields both WMMA B VGPRs {row kk+h, row kk+h+2}.
// ---------------------------------------------------------------------------
__global__ void wprep_kernel(const float* __restrict__ W, v2f* __restrict__ WP) {
  int i = blockIdx.x * blockDim.x + threadIdx.x;
  if (i < 256 * 128) {
    int k = i >> 7;
    int o = i & 127;
    v2f p;
    p.x = W[o * 256 + k];
    p.y = W[o * 256 + ((k + 2) & 255)];
    WP[i] = p;
  }
}

// ---------------------------------------------------------------------------
// One fused GraphSAGE layer for a group of 32 rows:
//   out[row] = relu( concat(self[srow], mean_j neigh[nrow_j]) @ W^T )
// GATHER=false -> identity maps (layer 0), GATHER=true -> frontier gathers.
// Block = 256 threads = 8 waves; wave w owns output columns [16w,16w+16) for
// both node halves (two v8f accumulators sharing each B fragment):
//   per K-step: 2x ds_load_b64 (A) + 1x global_load_b64 (B) + 2x wmma.
// ---------------------------------------------------------------------------
template <bool GATHER>
__global__ __launch_bounds__(256)
void sage_layer_kernel(const float* __restrict__ self_f,
                       const float* __restrict__ neigh_f,
                       const int*   __restrict__ smap,
                       const int*   __restrict__ nmap,
                       const v2f*   __restrict__ WP,     // [256][128] float2 pairs
                       float*       __restrict__ outp) { // [nrows][128]
  __shared__ float comb[NODES * CSTR];   // node-major: comb[n][k], k in [0,256)

  const int t = threadIdx.x;
  const int g = blockIdx.x;

  // ---- Phase 1: build combined rows in LDS (b128 in, b128 out) ------------
  // 1024 (n, f4) items over 256 threads; a wave covers one 512B feature row.
  // n = oid>>5 is wave-uniform -> gather indices scalarize.
  #pragma unroll
  for (int r = 0; r < 4; ++r) {
    const int oid = t + 256 * r;        // 0..1023
    const int n   = oid >> 5;           // node within group (0..31)
    const int f   = (oid & 31) * 4;     // feature chunk (0,4,...,124)
    const int row = g * NODES + n;
    const int srow = GATHER ? smap[row] : row;

    *(v4f*)&comb[n * CSTR + f] =
        *(const v4f*)(self_f + (long)srow * D_IN + f);

    v4f s = {};
    #pragma unroll
    for (int j = 0; j < K_NEI; ++j) {
      const int nr = GATHER ? nmap[row * K_NEI + j] : (row * K_NEI + j);
      s += *(const v4f*)(neigh_f + (long)nr * D_IN + f);
    }
    *(v4f*)&comb[n * CSTR + 128 + f] = s * (1.0f / (float)K_NEI);
  }
  __syncthreads();

  // ---- Phase 2: WMMA f32 16x16x4, K = 256, two node-half tiles ------------
  const int wave = t >> 5;
  const int lane = t & 31;
  const int l15  = lane & 15;
  const int h    = lane >> 4;          // half-wave select
  const int o0   = wave * 16;          // this wave's output-column block

  // A-frag: lanes 0-15 -> {K=kk, kk+1}; lanes 16-31 -> {K=kk+2, kk+3};
  // contiguous in node-major comb -> single ds_load_b64 per tile.
  const float* cb = comb + l15 * CSTR + 2 * h;
  // B-frag: VGPR0 = row kk+h, VGPR1 = row kk+h+2 -> one global_load_b64.
  const v2f* wp = WP + (long)h * 128 + o0 + l15;

  v8f acc0 = {};
  v8f acc1 = {};
  #pragma unroll 8
  for (int kk = 0; kk < KDIM; kk += 4) {
    const v2f a0 = *(const v2f*)(cb + kk);                // nodes 0..15
    const v2f a1 = *(const v2f*)(cb + 16 * CSTR + kk);    // nodes 16..31
    const v2f b  = wp[(long)kk * 128];
    acc0 = __builtin_amdgcn_wmma_f32_16x16x4_f32(
        false, a0, false, b, (short)0, acc0, false, false);
    acc1 = __builtin_amdgcn_wmma_f32_16x16x4_f32(
        false, a1, false, b, (short)0, acc1, false, false);
  }

  // ---- Phase 3: fused ReLU + coalesced stores -----------------------------
  // D layout: VGPR r, lanes 0-15 -> node=r, out=o0+l15; lanes 16-31 -> node=8+r
  float* op0 = outp + (long)(g * NODES +      8 * h) * D_IN + o0 + l15;
  float* op1 = outp + (long)(g * NODES + 16 + 8 * h) * D_IN + o0 + l15;
  #pragma unroll
  for (int r = 0; r < 8; ++r) {
    const float v0 = acc0[r];
    const float v1 = acc1[r];
    op0[(long)r * D_IN] = v0 > 0.f ? v0 : 0.f;
    op1[(long)r * D_IN] = v1 > 0.f ? v1 : 0.f;
  }
}

// ---------------------------------------------------------------------------
// scores[b, c] = dot(center[b], cn[b*C + c]) over 128 floats.
// One wave per output; float4 loads; 5-step butterfly reduction.
// ---------------------------------------------------------------------------
__global__ __launch_bounds__(256)
void scores_kernel(const float* __restrict__ center,
                   const float* __restrict__ cn,
                   float* __restrict__ scores, int total, int C) {
  const int gw   = (blockIdx.x * blockDim.x + threadIdx.x) >> 5;
  const int lane = threadIdx.x & 31;
  if (gw >= total) return;                 // uniform per-wave
  const int b = gw / C;
  const float4 x = ((const float4*)(center + (long)b  * 128))[lane];
  const float4 y = ((const float4*)(cn     + (long)gw * 128))[lane];
  float s = x.x * y.x + x.y * y.y + x.z * y.z + x.w * y.w;
  #pragma unroll
  for (int off = 16; off; off >>= 1) s += __shfl_xor(s, off, 32);
  if (lane == 0) scores[gw] = s;
}

// ---------------------------------------------------------------------------
extern "C" void kernel_launch(void* const* d_in, const int* in_sizes, int n_in,
                              void* d_out, int out_size, void* d_ws, size_t ws_size,
                              hipStream_t stream) {
  const float* c_self   = (const float*)d_in[0];   // [N0C, 128]
  const int*   c_map    = (const int*)  d_in[1];   // [B]
  const float* c_neigh  = (const float*)d_in[2];   // [N0C, 16, 128]
  const int*   c_nmap   = (const int*)  d_in[3];   // [B, 16]
  const float* cn_self  = (const float*)d_in[4];   // [N0CN, 128]
  const int*   cn_map   = (const int*)  d_in[5];   // [B*C]
  const float* cn_neigh = (const float*)d_in[6];   // [N0CN, 16, 128]
  const int*   cn_nmap  = (const int*)  d_in[7];   // [B*C, 16]
  const float* W0       = (const float*)d_in[10];  // [128, 256]
  const float* W1       = (const float*)d_in[11];  // [128, 256]

  const int n0c  = in_sizes[0] / D_IN;   // 20000
  const int bsz  = in_sizes[1];          // 4096
  const int n0cn = in_sizes[4] / D_IN;   // 60000
  const int bc   = in_sizes[5];          // 24576
  const int C    = bc / bsz;             // 6

  // Workspace layout.
  char* ws = (char*)d_ws;
  size_t off = 0;
  v2f*   WP0  = (v2f*)  (ws + off); off += 256 * 128 * sizeof(v2f);
  v2f*   WP1  = (v2f*)  (ws + off); off += 256 * 128 * sizeof(v2f);
  float* h0c  = (float*)(ws + off); off += (size_t)n0c  * D_IN * sizeof(float);
  float* h0cn = (float*)(ws + off); off += (size_t)n0cn * D_IN * sizeof(float);
  float* cnh1 = (float*)(ws + off); off += (size_t)bc   * D_IN * sizeof(float);

  float* center_out = (float*)d_out;                       // [B, 128]
  float* scores_out = (float*)d_out + (size_t)bsz * D_IN;  // [B, 1, C]

  // 1) Pre-build pair-interleaved transposed weights (tiny; L2-resident).
  wprep_kernel<<<128, 256, 0, stream>>>(W0, WP0);
  wprep_kernel<<<128, 256, 0, stream>>>(W1, WP1);

  // 2) Layer 0 over all nodes (identity maps).
  sage_layer_kernel<false><<<n0c  / NODES, 256, 0, stream>>>(
      c_self,  c_neigh,  nullptr, nullptr, WP0, h0c);
  sage_layer_kernel<false><<<n0cn / NODES, 256, 0, stream>>>(
      cn_self, cn_neigh, nullptr, nullptr, WP0, h0cn);

  // 3) Layer 1 with frontier gathers (self and neigh both gather from h0).
  sage_layer_kernel<true><<<bsz / NODES, 256, 0, stream>>>(
      h0c,  h0c,  c_map,  c_nmap,  WP1, center_out);
  sage_layer_kernel<true><<<bc  / NODES, 256, 0, stream>>>(
      h0cn, h0cn, cn_map, cn_nmap, WP1, cnh1);

  // 4) Scores: one wave per (b, c).
  const int total  = bc;                      // B*C dots
  const int blocks = (total * 32 + 255) / 256;
  scores_kernel<<<blocks, 256, 0, stream>>>(center_out, cnh1, scores_out, total, C);
}